// LightGCNConv_27642409517744
// MI455X (gfx1250) — compile-verified
//
#include <hip/hip_runtime.h>

// LightGCN conv: out[row] += x[col] * w  over E edges, DIM=64.
// Memory-bound scatter-add. Strategy for MI455X (gfx1250):
//  - 16 lanes per edge, float4 (b128) gather of x[col], coalesced 256B/edge
//  - native global_atomic_add_f32 (no-return) into out[row]; out + x are
//    L2-resident (51 MB << 192 MB L2), so atomics resolve in L2
//  - edge stream loaded non-temporally to avoid polluting L2

#ifndef DIMV
#define DIMV 64
#endif

__global__ __launch_bounds__(256) void lightgcn_scatter_kernel(
    const float* __restrict__ x,        // [N, 64]
    const int*   __restrict__ eidx,     // [2, E] (row-major: rows then cols)
    const float* __restrict__ ew,       // [E]
    float*       __restrict__ out,      // [N, 64]
    int n_edges)
{
    const int t = blockIdx.x * blockDim.x + threadIdx.x;
    const int e = t >> 4;        // 16 lanes cooperate on one edge
    const int q = t & 15;        // which float4 chunk of the 64-wide row
    if (e >= n_edges) return;

    // Streaming, touched-once edge data: non-temporal loads (th:TH_LOAD_NT)
    const int   row = __builtin_nontemporal_load(eidx + e);
    const int   col = __builtin_nontemporal_load(eidx + n_edges + e);
    const float w   = __builtin_nontemporal_load(ew + e);

    // Coalesced 256B gather of x[col]: one global_load_b128 per lane
    const float4 v = *(reinterpret_cast<const float4*>(x + (size_t)col * DIMV) + q);

    // Scatter-add: native global_atomic_add_f32 (no return), resolves in L2
    float* o = out + (size_t)row * DIMV + (size_t)q * 4;
    unsafeAtomicAdd(o + 0, v.x * w);
    unsafeAtomicAdd(o + 1, v.y * w);
    unsafeAtomicAdd(o + 2, v.z * w);
    unsafeAtomicAdd(o + 3, v.w * w);
}

extern "C" void kernel_launch(void* const* d_in, const int* in_sizes, int n_in,
                              void* d_out, int out_size, void* d_ws, size_t ws_size,
                              hipStream_t stream) {
    const float* x    = (const float*)d_in[0];   // [N, 64] f32
    const int*   eidx = (const int*)  d_in[1];   // [2, E]  int
    const float* ew   = (const float*)d_in[2];   // [E]     f32
    float*       out  = (float*)d_out;           // [N, 64] f32

    const int n_edges = in_sizes[2];             // E from edge_weight

    // Kernel accumulates; harness poisons d_out -> zero it every call.
    hipMemsetAsync(out, 0, (size_t)out_size * sizeof(float), stream);

    const int threads = 256;                      // 8 wave32 per block
    const long long total = (long long)n_edges * 16;
    const int blocks = (int)((total + threads - 1) / threads);

    lightgcn_scatter_kernel<<<blocks, threads, 0, stream>>>(x, eidx, ew, out, n_edges);
}